// DiffPoolModel_30391188587265
// MI455X (gfx1250) — compile-verified
//
#include <hip/hip_runtime.h>
#include <hip/hip_bf16.h>

typedef __attribute__((ext_vector_type(2))) float v2f;
typedef __attribute__((ext_vector_type(8))) float v8f;

// ---------------------------------------------------------------------------
// Neighbor aggregation: one block per graph, LDS accumulation via ds_add_f32.
// out[n,f] = sum_{e: row==n} x[col[e], f] * ew[e]   (optionally / max(cnt,1))
// dense!=0 : implicit dense MPG x MPG graph (row=e/MPG, col=e%MPG).
// Feature range split over blockIdx.y in chunks of 64 to keep LDS <= 34 KB.
// ---------------------------------------------------------------------------
__global__ __launch_bounds__(256)
void agg_kernel(const float* __restrict__ x,
                const int* __restrict__ row, const int* __restrict__ col,
                const float* __restrict__ ew,
                float* __restrict__ out,
                float* __restrict__ cnt_out, float* __restrict__ deg_out,
                int MPG, int EPG, int F, int do_mean, int dense) {
  extern __shared__ float lds[];
  const int g  = blockIdx.x;
  const int f0 = blockIdx.y * 64;
  const int FC = (F - f0) < 64 ? (F - f0) : 64;
  float* acc = lds;                // MPG*FC
  float* cnt = lds + MPG * FC;     // MPG
  float* dg  = cnt + MPG;          // MPG
  const int tid = threadIdx.x;
  for (int i = tid; i < MPG * FC; i += 256) acc[i] = 0.f;
  for (int i = tid; i < 2 * MPG;  i += 256) cnt[i] = 0.f;  // cnt+dg contiguous
  __syncthreads();

  const int ebase = g * EPG, nbase = g * MPG;
  const int lane = tid & 31, wave = tid >> 5;
  for (int e = wave; e < EPG; e += 8) {
    int r, c;
    if (dense) { r = e / MPG; c = e % MPG; }
    else       { r = row[ebase + e] - nbase; c = col[ebase + e] - nbase; }
    const float w = ew[ebase + e];
    if (lane == 0) { atomicAdd(&cnt[r], 1.0f); atomicAdd(&dg[r], w); }
    const float* xr = x + (nbase + c) * F + f0;
    for (int f = lane; f < FC; f += 32)
      atomicAdd(&acc[r * FC + f], xr[f] * w);
  }
  __syncthreads();

  for (int i = tid; i < MPG * FC; i += 256) {
    const int n = i / FC, f = i % FC;
    float v = acc[i];
    if (do_mean) v /= fmaxf(cnt[n], 1.0f);
    out[(nbase + n) * F + f0 + f] = v;
  }
  if (blockIdx.y == 0 && cnt_out != nullptr) {
    for (int i = tid; i < MPG; i += 256) {
      cnt_out[nbase + i] = cnt[i];
      deg_out[nbase + i] = dg[i];
    }
  }
}

// ---------------------------------------------------------------------------
// SAGE GEMM via V_WMMA_F32_16X16X4_F32 (fp32 matrix core path).
// out = act(X @ Ws + Mn @ Wn + bias),  X,Mn: [N,Din], W: [Din,F], out: [N,F].
// Block = 128 threads (4 waves); each wave: one 16-col tile x four 16-row
// tiles (64 rows/block). Manually double-buffered: K-step k+1's 10 fragment
// loads are issued before K-step k's 8 WMMAs so loads overlap matrix math
// (distinct live ranges defeat the reg-recycling load->wait->wmma serial
// pattern). Out-of-range columns: address-clamped loads, garbage only lives
// in lanes whose stores are predicated off. grid = (N/64, ceil(F/64)).
// ---------------------------------------------------------------------------
__global__ __launch_bounds__(128)
void sage_gemm(const float* __restrict__ X, const float* __restrict__ Mn,
               const float* __restrict__ Ws, const float* __restrict__ Wn,
               const float* __restrict__ bias, float* __restrict__ out,
               int N, int Din, int F, int act) {
  const int wave = threadIdx.x >> 5;
  const int lane = threadIdx.x & 31;
  const int half = lane >> 4;      // K-pair selector for A/B fragments
  const int lr   = lane & 15;
  const int rowBase = blockIdx.x * 64;
  const int colA = (blockIdx.y * 4 + wave) * 16 + lr;
  const bool colOK = colA < F;
  const int colC = colOK ? colA : 0;          // clamped: loads always legal

  const int rstep = 16 * Din;                 // floats between row tiles
  const float* xp  = X  + (rowBase + lr) * Din + 2 * half;  // advances 4/iter
  const float* mp  = Mn + (rowBase + lr) * Din + 2 * half;
  const float* wsp = Ws + 2 * half * F + colC;              // advances 4F/iter
  const float* wnp = Wn + 2 * half * F + colC;

  v8f c0 = {0.f,0.f,0.f,0.f,0.f,0.f,0.f,0.f};
  v8f c1 = c0, c2 = c0, c3 = c0;

  // ---- preload K-step 0 fragments ----
  v2f b, b2, a0, a1, a2, a3, m0, m1, m2, m3;
  b.x  = wsp[0];  b.y  = wsp[F];
  b2.x = wnp[0];  b2.y = wnp[F];
  a0 = *(const v2f*)(xp);
  a1 = *(const v2f*)(xp + rstep);
  a2 = *(const v2f*)(xp + 2 * rstep);
  a3 = *(const v2f*)(xp + 3 * rstep);
  m0 = *(const v2f*)(mp);
  m1 = *(const v2f*)(mp + rstep);
  m2 = *(const v2f*)(mp + 2 * rstep);
  m3 = *(const v2f*)(mp + 3 * rstep);

  // ---- steady state: issue next step's loads, then current step's WMMAs ----
  for (int k0 = 0; k0 < Din - 4; k0 += 4) {
    xp += 4; mp += 4; wsp += 4 * F; wnp += 4 * F;
    v2f nb, nb2;
    nb.x  = wsp[0];  nb.y  = wsp[F];
    nb2.x = wnp[0];  nb2.y = wnp[F];
    const v2f na0 = *(const v2f*)(xp);
    const v2f na1 = *(const v2f*)(xp + rstep);
    const v2f na2 = *(const v2f*)(xp + 2 * rstep);
    const v2f na3 = *(const v2f*)(xp + 3 * rstep);
    const v2f nm0 = *(const v2f*)(mp);
    const v2f nm1 = *(const v2f*)(mp + rstep);
    const v2f nm2 = *(const v2f*)(mp + 2 * rstep);
    const v2f nm3 = *(const v2f*)(mp + 3 * rstep);

    c0 = __builtin_amdgcn_wmma_f32_16x16x4_f32(false, a0, false, b,  (short)0, c0, false, false);
    c1 = __builtin_amdgcn_wmma_f32_16x16x4_f32(false, a1, false, b,  (short)0, c1, false, false);
    c2 = __builtin_amdgcn_wmma_f32_16x16x4_f32(false, a2, false, b,  (short)0, c2, false, false);
    c3 = __builtin_amdgcn_wmma_f32_16x16x4_f32(false, a3, false, b,  (short)0, c3, false, false);
    c0 = __builtin_amdgcn_wmma_f32_16x16x4_f32(false, m0, false, b2, (short)0, c0, false, false);
    c1 = __builtin_amdgcn_wmma_f32_16x16x4_f32(false, m1, false, b2, (short)0, c1, false, false);
    c2 = __builtin_amdgcn_wmma_f32_16x16x4_f32(false, m2, false, b2, (short)0, c2, false, false);
    c3 = __builtin_amdgcn_wmma_f32_16x16x4_f32(false, m3, false, b2, (short)0, c3, false, false);

    b = nb; b2 = nb2;
    a0 = na0; a1 = na1; a2 = na2; a3 = na3;
    m0 = nm0; m1 = nm1; m2 = nm2; m3 = nm3;
  }

  // ---- peeled final K-step ----
  c0 = __builtin_amdgcn_wmma_f32_16x16x4_f32(false, a0, false, b,  (short)0, c0, false, false);
  c1 = __builtin_amdgcn_wmma_f32_16x16x4_f32(false, a1, false, b,  (short)0, c1, false, false);
  c2 = __builtin_amdgcn_wmma_f32_16x16x4_f32(false, a2, false, b,  (short)0, c2, false, false);
  c3 = __builtin_amdgcn_wmma_f32_16x16x4_f32(false, a3, false, b,  (short)0, c3, false, false);
  c0 = __builtin_amdgcn_wmma_f32_16x16x4_f32(false, m0, false, b2, (short)0, c0, false, false);
  c1 = __builtin_amdgcn_wmma_f32_16x16x4_f32(false, m1, false, b2, (short)0, c1, false, false);
  c2 = __builtin_amdgcn_wmma_f32_16x16x4_f32(false, m2, false, b2, (short)0, c2, false, false);
  c3 = __builtin_amdgcn_wmma_f32_16x16x4_f32(false, m3, false, b2, (short)0, c3, false, false);

  if (!colOK) return;
  const float bv = bias[colA];
  v8f cc[4] = {c0, c1, c2, c3};
  #pragma unroll
  for (int rt = 0; rt < 4; rt++) {
    #pragma unroll
    for (int r = 0; r < 8; r++) {
      const int m = rowBase + rt * 16 + r + 8 * half;  // C/D layout
      float v = cc[rt][r] + bv;
      if (act) v = fmaxf(v, 0.f);
      out[m * F + colA] = v;
    }
  }
}

// ---------------------------------------------------------------------------
// Row-wise softmax (K <= 20), one thread per node, in place.
// ---------------------------------------------------------------------------
__global__ __launch_bounds__(256)
void softmax_rows(float* __restrict__ s, int N, int K) {
  const int n = blockIdx.x * blockDim.x + threadIdx.x;
  if (n >= N) return;
  float* r = s + n * K;
  float mx = -1e30f;
  for (int k = 0; k < K; k++) mx = fmaxf(mx, r[k]);
  float sum = 0.f;
  for (int k = 0; k < K; k++) { float e = __expf(r[k] - mx); r[k] = e; sum += e; }
  const float inv = 1.f / sum;
  for (int k = 0; k < K; k++) r[k] *= inv;
}

// ---------------------------------------------------------------------------
// Per-graph pooling + mincut losses. One block per graph.
// ---------------------------------------------------------------------------
__global__ __launch_bounds__(256)
void pool_kernel(const float* __restrict__ S, const float* __restrict__ H,
                 const float* __restrict__ AS, const float* __restrict__ deg,
                 const float* __restrict__ pbias,
                 float* __restrict__ pooled_x, float* __restrict__ SAS_out,
                 float* __restrict__ gmax, float* __restrict__ cut_arr,
                 float* __restrict__ orth_arr, int M, int K, int F) {
  extern __shared__ float lds[];
  float* sS   = lds;               // M*K
  float* sAS  = sS   + M * K;      // M*K
  float* sPX  = sAS  + M * K;      // K*F
  float* sSAS = sPX  + K * F;      // K*K
  float* sStS = sSAS + K * K;      // K*K
  float* sRed = sStS + K * K;      // 256
  const int g = blockIdx.x, tid = threadIdx.x;
  const int nb = g * M;

  for (int i = tid; i < M * K; i += 256) { sS[i] = S[nb * K + i]; sAS[i] = AS[nb * K + i]; }
  __syncthreads();

  for (int i = tid; i < K * F; i += 256) {
    const int k = i / F, f = i % F;
    float acc = 0.f;
    for (int m = 0; m < M; m++) acc += sS[m * K + k] * H[(nb + m) * F + f];
    const float v = fmaxf(acc + pbias[f], 0.f);
    sPX[i] = v;
    pooled_x[g * K * F + i] = v;
  }
  for (int i = tid; i < K * K; i += 256) {
    const int k = i / K, k2 = i % K;
    float a = 0.f, b = 0.f;
    for (int m = 0; m < M; m++) {
      const float sm = sS[m * K + k];
      a += sm * sAS[m * K + k2];
      b += sm * sS[m * K + k2];
    }
    sSAS[i] = a; sStS[i] = b;
    SAS_out[g * K * K + i] = a;
  }
  __syncthreads();

  for (int f = tid; f < F; f += 256) {
    float mx = -1e30f;
    for (int k = 0; k < K; k++) mx = fmaxf(mx, sPX[k * F + f]);
    gmax[g * F + f] = mx;
  }

  float part = 0.f;
  for (int m = tid; m < M; m += 256) {
    float ss = 0.f;
    for (int k = 0; k < K; k++) { const float v = sS[m * K + k]; ss += v * v; }
    part += deg[nb + m] * ss;
  }
  sRed[tid] = part; __syncthreads();
  for (int s = 128; s > 0; s >>= 1) { if (tid < s) sRed[tid] += sRed[tid + s]; __syncthreads(); }
  const float cut_den = sRed[0];
  __syncthreads();

  part = 0.f;
  for (int i = tid; i < K * K; i += 256) part += sStS[i] * sStS[i];
  sRed[tid] = part; __syncthreads();
  for (int s = 128; s > 0; s >>= 1) { if (tid < s) sRed[tid] += sRed[tid + s]; __syncthreads(); }
  const float fro = sqrtf(sRed[0] + 1e-9f);
  __syncthreads();

  const float invsk = rsqrtf((float)K);
  part = 0.f;
  for (int i = tid; i < K * K; i += 256) {
    const float e = ((i / K) == (i % K)) ? invsk : 0.f;
    const float d = sStS[i] / fro - e;
    part += d * d;
  }
  sRed[tid] = part; __syncthreads();
  for (int s = 128; s > 0; s >>= 1) { if (tid < s) sRed[tid] += sRed[tid + s]; __syncthreads(); }
  if (tid == 0) {
    float cut_num = 0.f;
    for (int k = 0; k < K; k++) cut_num += sSAS[k * K + k];
    cut_arr[g]  = -cut_num / (cut_den + 1e-9f);
    orth_arr[g] = sqrtf(sRed[0] + 1e-9f);
  }
}

// ---------------------------------------------------------------------------
// Final MLP on concat(gmax1, gmax2) + loss reduction. One 256-thread block.
// ---------------------------------------------------------------------------
__global__ __launch_bounds__(256)
void final_kernel(const float* __restrict__ g1, const float* __restrict__ g2,
                  const float* __restrict__ W, const float* __restrict__ b,
                  const float* __restrict__ c1, const float* __restrict__ o1,
                  const float* __restrict__ c2, const float* __restrict__ o2,
                  float* __restrict__ out, int G, int F) {
  __shared__ float red[256];
  const int tid = threadIdx.x;
  for (int idx = tid; idx < G * 2; idx += 256) {
    const int g = idx >> 1, cc = idx & 1;
    float acc = b[cc];
    for (int f = 0; f < F; f++) acc += g1[g * F + f] * W[f * 2 + cc];
    for (int f = 0; f < F; f++) acc += g2[g * F + f] * W[(F + f) * 2 + cc];
    out[idx] = acc;
  }
  float pc = (tid < G) ? (c1[tid] + c2[tid]) : 0.f;
  float po = (tid < G) ? (o1[tid] + o2[tid]) : 0.f;
  red[tid] = pc; __syncthreads();
  for (int s = 128; s > 0; s >>= 1) { if (tid < s) red[tid] += red[tid + s]; __syncthreads(); }
  if (tid == 0) out[G * 2] = red[0] / (float)G;
  __syncthreads();
  red[tid] = po; __syncthreads();
  for (int s = 128; s > 0; s >>= 1) { if (tid < s) red[tid] += red[tid + s]; __syncthreads(); }
  if (tid == 0) out[G * 2 + 1] = red[0] / (float)G;
}

// ---------------------------------------------------------------------------
// Orchestration
// ---------------------------------------------------------------------------
extern "C" void kernel_launch(void* const* d_in, const int* in_sizes, int n_in,
                              void* d_out, int out_size, void* d_ws, size_t ws_size,
                              hipStream_t stream) {
  constexpr int G = 256, M1 = 128, E1 = 2048, F = 128, K1 = 20;
  constexpr int N1 = G * M1;                 // 32768
  constexpr int M2 = K1, E2 = K1 * K1, K2 = 5;
  constexpr int N2 = G * M2;                 // 5120

  const float* x   = (const float*)d_in[0];
  const int*   row = (const int*)d_in[1];
  const int*   col = row + (size_t)G * E1;
  const float* ew  = (const float*)d_in[2];
  auto P = [&](int i) { return (const float*)d_in[i]; };
  // jax dict-sorted flatten: per level [assign0{s,n,b}, assign1{s,n,b},
  // feat0{s,n,b}, feat1{s,n,b}, pool_bias]; then mlp_b, mlp_w.
  // L0: 4..16 ; L1: 17..29 ; 30 mlp_b ; 31 mlp_w

  float* ws = (float*)d_ws;
  const size_t SLOT = (size_t)N1 * F;
  float* s0 = ws;            float* s1 = s0 + SLOT;  float* s2 = s1 + SLOT;
  float* s3 = s2 + SLOT;     float* s4 = s3 + SLOT;  float* s5 = s4 + SLOT;
  float* S1   = s5 + SLOT;                 // N1*K1
  float* AS1  = S1  + (size_t)N1 * K1;
  float* PX1  = AS1 + (size_t)N1 * K1;     // G*K1*F
  float* SAS1 = PX1 + (size_t)G * K1 * F;  // G*K1*K1
  float* cnt1 = SAS1 + (size_t)G * K1 * K1;
  float* deg1 = cnt1 + N1;
  float* gmx1 = deg1 + N1;                 // G*F
  float* gmx2 = gmx1 + (size_t)G * F;
  float* S2   = gmx2 + (size_t)G * F;      // N2*K2
  float* AS2  = S2  + (size_t)N2 * K2;
  float* PX2  = AS2 + (size_t)N2 * K2;     // G*K2*F
  float* SAS2 = PX2 + (size_t)G * K2 * F;
  float* cnt2 = SAS2 + (size_t)G * K2 * K2;
  float* deg2 = cnt2 + N2;
  float* c1a  = deg2 + N2;
  float* o1a  = c1a + G;
  float* c2a  = o1a + G;
  float* o2a  = c2a + G;

  auto aggLds = [](int M, int Fc) { return (size_t)(M * (Fc < 64 ? Fc : 64) + 2 * M) * 4; };

  // ---------------- Level 1 (Din=64, K=20) ----------------
  agg_kernel<<<dim3(G, 1), 256, aggLds(M1, 64), stream>>>(x, row, col, ew, s0, cnt1, deg1, M1, E1, 64, 1, 0);
  sage_gemm<<<dim3(N1 / 64, 2), 128, 0, stream>>>(x, s0, P(10), P(11), P(12), s1, N1, 64, F, 1); // feat0
  sage_gemm<<<dim3(N1 / 64, 2), 128, 0, stream>>>(x, s0, P(4),  P(5),  P(6),  s2, N1, 64, F, 1); // assign0
  agg_kernel<<<dim3(G, 2), 256, aggLds(M1, F), stream>>>(s1, row, col, ew, s3, nullptr, nullptr, M1, E1, F, 1, 0);
  agg_kernel<<<dim3(G, 2), 256, aggLds(M1, F), stream>>>(s2, row, col, ew, s4, nullptr, nullptr, M1, E1, F, 1, 0);
  sage_gemm<<<dim3(N1 / 64, 2), 128, 0, stream>>>(s1, s3, P(13), P(14), P(15), s5, N1, F, F, 0);  // feat1 -> h
  sage_gemm<<<dim3(N1 / 64, 1), 128, 0, stream>>>(s2, s4, P(7),  P(8),  P(9),  S1, N1, F, K1, 0); // assign1
  softmax_rows<<<N1 / 256, 256, 0, stream>>>(S1, N1, K1);
  agg_kernel<<<dim3(G, 1), 256, aggLds(M1, K1), stream>>>(S1, row, col, ew, AS1, nullptr, nullptr, M1, E1, K1, 0, 0);
  {
    const size_t lds = (size_t)(2 * M1 * K1 + K1 * F + 2 * K1 * K1 + 256) * 4;
    pool_kernel<<<G, 256, lds, stream>>>(S1, s5, AS1, deg1, P(16), PX1, SAS1, gmx1, c1a, o1a, M1, K1, F);
  }

  // ---------------- Level 2 (Din=128, K=5), dense 20x20 graphs, ew = SAS1 ----------------
  agg_kernel<<<dim3(G, 2), 256, aggLds(M2, F), stream>>>(PX1, nullptr, nullptr, SAS1, s0, cnt2, deg2, M2, E2, F, 1, 1);
  sage_gemm<<<dim3(N2 / 64, 2), 128, 0, stream>>>(PX1, s0, P(23), P(24), P(25), s1, N2, F, F, 1); // feat0
  sage_gemm<<<dim3(N2 / 64, 2), 128, 0, stream>>>(PX1, s0, P(17), P(18), P(19), s2, N2, F, F, 1); // assign0
  agg_kernel<<<dim3(G, 2), 256, aggLds(M2, F), stream>>>(s1, nullptr, nullptr, SAS1, s3, nullptr, nullptr, M2, E2, F, 1, 1);
  agg_kernel<<<dim3(G, 2), 256, aggLds(M2, F), stream>>>(s2, nullptr, nullptr, SAS1, s4, nullptr, nullptr, M2, E2, F, 1, 1);
  sage_gemm<<<dim3(N2 / 64, 2), 128, 0, stream>>>(s1, s3, P(26), P(27), P(28), s5, N2, F, F, 0);  // feat1 -> h2
  sage_gemm<<<dim3(N2 / 64, 1), 128, 0, stream>>>(s2, s4, P(20), P(21), P(22), S2, N2, F, K2, 0); // assign1
  softmax_rows<<<N2 / 256, 256, 0, stream>>>(S2, N2, K2);
  agg_kernel<<<dim3(G, 1), 256, aggLds(M2, K2), stream>>>(S2, nullptr, nullptr, SAS1, AS2, nullptr, nullptr, M2, E2, K2, 0, 1);
  {
    const size_t lds = (size_t)(2 * M2 * K2 + K2 * F + 2 * K2 * K2 + 256) * 4;
    pool_kernel<<<G, 256, lds, stream>>>(S2, s5, AS2, deg2, P(29), PX2, SAS2, gmx2, c2a, o2a, M2, K2, F);
  }

  // ---------------- Readout ----------------
  final_kernel<<<1, 256, 0, stream>>>(gmx1, gmx2, P(31), P(30), c1a, o1a, c2a, o2a, (float*)d_out, G, F);
}